// GNN_15247133901709
// MI455X (gfx1250) — compile-verified
//
#include <hip/hip_runtime.h>
#include <hip/hip_bf16.h>

// ---------------------------------------------------------------------------
// GCN forward (2x GCNConv + mean-pool + FC + L2 normalize) for MI455X gfx1250.
// Dense GEMMs use V_WMMA_F32_16X16X4_F32 (fp32 matrix pipe -> matches the
// fp32 reference numerically). The GEMM is persistent: each wave hoists its
// 16 B(weight) fragments into registers once and grid-strides over row tiles,
// so the steady-state loop is A-loads + WMMA + straight-line stores.
// Scatter/aggregation uses global f32 atomics; all intermediate buffers
// (~26 MB) fit in the 192 MB L2.
// ---------------------------------------------------------------------------

typedef __attribute__((ext_vector_type(2))) float v2f;
typedef __attribute__((ext_vector_type(8))) float v8f;

#define F_DIM 64  // F_IN == H == D_EMB == 64

// ---------------- elementwise / setup kernels ------------------------------

__global__ void k_zero(float* __restrict__ p, long long n) {
  long long i = blockIdx.x * (long long)blockDim.x + threadIdx.x;
  long long stride = (long long)gridDim.x * blockDim.x;
  for (; i < n; i += stride) p[i] = 0.0f;
}

__global__ void k_deg(const int* __restrict__ ei, int E, float* __restrict__ deg) {
  int e = blockIdx.x * blockDim.x + threadIdx.x;
  int stride = gridDim.x * blockDim.x;
  for (; e < E; e += stride) {
    int d = ei[E + e];  // dst row of edge_index
    atomicAdd(&deg[d], 1.0f);
  }
}

__global__ void k_counts(const int* __restrict__ batch, int N, float* __restrict__ counts) {
  int i = blockIdx.x * blockDim.x + threadIdx.x;
  int stride = gridDim.x * blockDim.x;
  for (; i < N; i += stride) atomicAdd(&counts[batch[i]], 1.0f);
}

__global__ void k_dinv(const float* __restrict__ deg, float* __restrict__ dinv, int N) {
  int i = blockIdx.x * blockDim.x + threadIdx.x;
  int stride = gridDim.x * blockDim.x;
  for (; i < N; i += stride) dinv[i] = rsqrtf(deg[i] + 1.0f);  // +1 self-loop
}

// ---------------- persistent WMMA fp32 GEMM --------------------------------
//   C[M x 64] = A[M x 64] * W[64 x 64]  (+ optional column bias)
// Block = 128 threads = 4 waves; wave w owns col tile [16w, 16w+16).
// B (weight) fragments for all 16 k-steps are loaded ONCE into 32 VGPRs,
// then the block grid-strides over 16-row strips: per tile only 16x
// global_load_b64 (A) + 16x V_WMMA_F32_16X16X4_F32 + 8 stores.
// Full tiles (the only kind for M % 16 == 0) take the unguarded store path.
// Fragment layouts (ISA 7.12.2, 32-bit):
//   A 16x4 : lanes 0-15 hold K={k,k+1} in VGPR0/1; lanes 16-31 K={k+2,k+3}
//   B 4x16 : row K striped across lanes within a VGPR (mirrored split)
//   C/D    : VGPR v -> M = v + 8*(lane>=16), N = lane%16

__global__ void __launch_bounds__(128)
k_gemm_wmma_f32(const float* __restrict__ A, const float* __restrict__ W,
                const float* __restrict__ bias, float* __restrict__ C,
                int M, int tiles) {
  const int wave = threadIdx.x >> 5;
  const int lane = threadIdx.x & 31;
  const int c0 = wave * 16;
  const int hi = lane >> 4;          // 0: lanes 0-15, 1: lanes 16-31
  const int nn = lane & 15;
  const int col = c0 + nn;           // B-frag / C column for this lane
  const float cbias = bias ? bias[col] : 0.0f;

  // hoist all 16 B fragments (loop-invariant weights) into registers
  v2f bf[16];
#pragma unroll
  for (int kk = 0; kk < 16; ++kk) {
    const int ka = kk * 4 + 2 * hi;
    bf[kk].x = W[(ka + 0) * F_DIM + col];
    bf[kk].y = W[(ka + 1) * F_DIM + col];
  }

  for (int t = blockIdx.x; t < tiles; t += gridDim.x) {
    const int r0 = t * 16;
    const float* ap = A + (long long)(r0 + nn) * F_DIM + 2 * hi;

    v8f acc = {};
#pragma unroll
    for (int kk = 0; kk < 16; ++kk) {
      v2f a;
      a.x = ap[kk * 4 + 0];          // contiguous float2 -> global_load_b64
      a.y = ap[kk * 4 + 1];
      acc = __builtin_amdgcn_wmma_f32_16x16x4_f32(
          /*neg_a=*/false, a, /*neg_b=*/false, bf[kk],
          /*c_mod=*/(short)0, acc, /*reuse_a=*/false, /*reuse_b=*/false);
    }

    float* cp = C + (long long)(r0 + 8 * hi) * F_DIM + col;
    if (r0 + 16 <= M) {
      // full tile: straight-line stores, no per-element exec masking
#pragma unroll
      for (int v = 0; v < 8; ++v) cp[(long long)v * F_DIM] = acc[v] + cbias;
    } else {
      // rare tail tile (never hit for M % 16 == 0)
#pragma unroll
      for (int v = 0; v < 8; ++v) {
        if (r0 + v + 8 * hi < M) cp[(long long)v * F_DIM] = acc[v] + cbias;
      }
    }
  }
}

// ---------------- edge scatter: agg[dst] += coef * feat[src] ----------------
// One thread per (edge, 4-feature group): float4 read + 4 f32 atomics (L2).

__global__ void k_scatter(const int* __restrict__ ei, int E,
                          const float* __restrict__ dinv,
                          const float* __restrict__ feat,
                          float* __restrict__ agg) {
  long long tid = blockIdx.x * (long long)blockDim.x + threadIdx.x;
  const long long total = (long long)E * (F_DIM / 4);
  const long long stride = (long long)gridDim.x * blockDim.x;
  for (; tid < total; tid += stride) {
    const int e = (int)(tid >> 4);
    const int f = (int)(tid & 15) * 4;
    const int s = ei[e];
    const int d = ei[E + e];
    const float coef = dinv[s] * dinv[d];
    const float4 v = *(const float4*)(feat + (long long)s * F_DIM + f);
    float* o = agg + (long long)d * F_DIM + f;
    atomicAdd(o + 0, coef * v.x);
    atomicAdd(o + 1, coef * v.y);
    atomicAdd(o + 2, coef * v.z);
    atomicAdd(o + 3, coef * v.w);
  }
}

// ---------------- finalize: agg = [relu](agg + dinv^2 * tmp + bias) ---------

__global__ void k_finalize(float* __restrict__ agg, const float* __restrict__ tmp,
                           const float* __restrict__ dinv, const float* __restrict__ bias,
                           int N, int do_relu) {
  long long idx = blockIdx.x * (long long)blockDim.x + threadIdx.x;
  const long long total = (long long)N * F_DIM;
  const long long stride = (long long)gridDim.x * blockDim.x;
  for (; idx < total; idx += stride) {
    const int i = (int)(idx >> 6);
    const int f = (int)(idx & 63);
    const float di = dinv[i];
    float v = agg[idx] + di * di * tmp[idx] + bias[f];
    if (do_relu) v = fmaxf(v, 0.0f);
    agg[idx] = v;
  }
}

// ---------------- pooling ---------------------------------------------------

__global__ void k_pool_add(const int* __restrict__ batch, const float* __restrict__ h,
                           float* __restrict__ sums, int N) {
  long long idx = blockIdx.x * (long long)blockDim.x + threadIdx.x;
  const long long total = (long long)N * F_DIM;
  const long long stride = (long long)gridDim.x * blockDim.x;
  for (; idx < total; idx += stride) {
    const int i = (int)(idx >> 6);
    const int f = (int)(idx & 63);
    atomicAdd(&sums[(long long)batch[i] * F_DIM + f], h[idx]);
  }
}

__global__ void k_pool_div(float* __restrict__ sums, const float* __restrict__ counts, int G) {
  int idx = blockIdx.x * blockDim.x + threadIdx.x;
  const int total = G * F_DIM;
  if (idx < total) {
    const int g = idx >> 6;
    sums[idx] = sums[idx] / fmaxf(counts[g], 1.0f);
  }
}

// ---------------- per-row L2 normalize (64 threads / row) -------------------

__global__ void k_l2norm(float* __restrict__ out) {
  __shared__ float red[F_DIM];
  const int g = blockIdx.x;
  const int t = threadIdx.x;
  const float v = out[(long long)g * F_DIM + t];
  red[t] = v * v;
  __syncthreads();
#pragma unroll
  for (int s = F_DIM / 2; s > 0; s >>= 1) {
    if (t < s) red[t] += red[t + s];
    __syncthreads();
  }
  const float nrm = fmaxf(sqrtf(red[0]), 1e-12f);
  out[(long long)g * F_DIM + t] = v / nrm;
}

// ---------------------------------------------------------------------------

extern "C" void kernel_launch(void* const* d_in, const int* in_sizes, int n_in,
                              void* d_out, int out_size, void* d_ws, size_t ws_size,
                              hipStream_t stream) {
  const float* x     = (const float*)d_in[0];
  const int*   ei    = (const int*)d_in[1];   // [2, E] flat: src row then dst row
  const int*   batch = (const int*)d_in[2];
  const float* W1    = (const float*)d_in[3];
  const float* b1    = (const float*)d_in[4];
  const float* W2    = (const float*)d_in[5];
  const float* b2    = (const float*)d_in[6];
  const float* Wfc   = (const float*)d_in[7];
  const float* bfc   = (const float*)d_in[8];
  float* out = (float*)d_out;

  const int N = in_sizes[2];          // 50000
  const int E = in_sizes[1] / 2;      // 800000
  const int G = out_size / F_DIM;     // 512

  // workspace partition (floats)
  float* ws     = (float*)d_ws;
  float* deg    = ws;                           // N
  float* counts = deg + N;                      // G
  float* dinv   = counts + G;                   // N
  float* bufA   = dinv + N;                     // N*64  (GEMM product)
  float* bufB   = bufA + (long long)N * F_DIM;  // N*64  (aggregation / layer out)
  float* sums   = bufB + (long long)N * F_DIM;  // G*64

  const int T = 256;
  const int gN   = (N + T - 1) / T;
  const int gE   = (E + T - 1) / T;
  const int gNF  = 2048;   // grid-stride over N*64 = 3.2M
  const int gSC  = 4096;   // grid-stride over E*16 = 12.8M

  const int tilesN = (N + 15) / 16;   // 3125 row strips
  const int tilesG = (G + 15) / 16;   // 32
  const int gGemmN = 1024;            // persistent blocks; ~3 tiles each

  // --- init ---------------------------------------------------------------
  k_zero<<<gN, T, 0, stream>>>(deg, N);
  k_zero<<<1, T, 0, stream>>>(counts, G);
  k_zero<<<gNF, T, 0, stream>>>(bufB, (long long)N * F_DIM);
  k_zero<<<(G * F_DIM + T - 1) / T, T, 0, stream>>>(sums, (long long)G * F_DIM);

  // --- degrees / graph counts / dinv --------------------------------------
  k_deg<<<gE, T, 0, stream>>>(ei, E, deg);
  k_counts<<<gN, T, 0, stream>>>(batch, N, counts);
  k_dinv<<<gN, T, 0, stream>>>(deg, dinv, N);

  // --- layer 1: h1 = relu(agg(x@W1) + b1) ---------------------------------
  k_gemm_wmma_f32<<<gGemmN, 128, 0, stream>>>(x, W1, nullptr, bufA, N, tilesN);
  k_scatter<<<gSC, T, 0, stream>>>(ei, E, dinv, bufA, bufB);
  k_finalize<<<gNF, T, 0, stream>>>(bufB, bufA, dinv, b1, N, /*relu=*/1);

  // --- layer 2: h2 = agg(h1@W2) + b2 --------------------------------------
  k_gemm_wmma_f32<<<gGemmN, 128, 0, stream>>>(bufB, W2, nullptr, bufA, N, tilesN);
  k_zero<<<gNF, T, 0, stream>>>(bufB, (long long)N * F_DIM);
  k_scatter<<<gSC, T, 0, stream>>>(ei, E, dinv, bufA, bufB);
  k_finalize<<<gNF, T, 0, stream>>>(bufB, bufA, dinv, b2, N, /*relu=*/0);

  // --- mean pool -----------------------------------------------------------
  k_pool_add<<<gNF, T, 0, stream>>>(batch, bufB, sums, N);
  k_pool_div<<<(G * F_DIM + T - 1) / T, T, 0, stream>>>(sums, counts, G);

  // --- FC + L2 normalize ---------------------------------------------------
  k_gemm_wmma_f32<<<tilesG, 128, 0, stream>>>(sums, Wfc, bfc, out, G, tilesG);
  k_l2norm<<<G, F_DIM, 0, stream>>>(out);
}